// CombinedLoss_73710228735023
// MI455X (gfx1250) — compile-verified
//
#include <hip/hip_runtime.h>

// Problem constants (pred: [16,5,512,512] f32, target: [16,512,512] i32)
#define NB   16
#define NC   5
#define NH   512
#define NW   512
#define ROWS 4
#define TPB  256
#define BIGD 2048   // 4*W, matches reference sentinel

typedef __attribute__((ext_vector_type(2))) float v2f;
typedef __attribute__((ext_vector_type(8))) float v8f;

// LDS layout: region X (logits 40KB, reused as 256x33 accumulator matrix 33.8KB),
// then target tile 8KB, then scan/dist tile 8KB. Total 56KB static LDS.
#define REGION_X   40960
#define OFF_TGT    (REGION_X)
#define OFF_SCAN   (REGION_X + 8192)
#define SMEM_BYTES (REGION_X + 8192 + 8192)

// ---- CDNA5 async global->LDS copy (ASYNCcnt-tracked, bypasses VGPRs) ----
__device__ __forceinline__ void async_copy_b128(const void* g, void* l) {
  unsigned lds = (unsigned)(unsigned long long)l;           // low 32 bits = LDS offset
  unsigned long long ga = (unsigned long long)g;
  asm volatile("global_load_async_to_lds_b128 %0, %1, off"
               :: "v"(lds), "v"(ga) : "memory");
}
__device__ __forceinline__ void wait_async_all() {
  asm volatile("s_wait_asynccnt 0x0" ::: "memory");
}

__global__ void __launch_bounds__(TPB) combined_loss_tile(
    const float* __restrict__ pred, const int* __restrict__ target,
    float* __restrict__ ws) {
  __shared__ __align__(16) unsigned char smem[SMEM_BYTES];
  float* s_logit = (float*)smem;                       // [NC][ROWS][NW]
  float* s_acc   = (float*)smem;                       // [TPB][33] (after barrier)
  int*   s_tgt   = (int*)(smem + OFF_TGT);             // [ROWS][NW]
  int*   s_scni  = (int*)(smem + OFF_SCAN);            // packed left indices
  float* s_scnf  = (float*)(smem + OFF_SCAN);          // signed distance (reused)

  const int tid  = threadIdx.x;
  const int bid  = blockIdx.x;
  const int b    = bid >> 7;           // /128 row-tiles per image
  const int row0 = (bid & 127) << 2;   // *ROWS

  // ---- stage tile into LDS asynchronously ----
  {
    const int* gt = target + ((size_t)b * NH + row0) * NW;
    for (int i = tid; i < (ROWS * NW) / 4; i += TPB)
      async_copy_b128(gt + i * 4, s_tgt + i * 4);
    for (int i = tid; i < (NC * ROWS * NW) / 4; i += TPB) {
      int e   = i * 4;
      int c   = e / (ROWS * NW);
      int rem = e - c * (ROWS * NW);   // r*NW + w
      const float* gp = pred + (((size_t)(b * NC + c) * NH + row0) * NW) + rem;
      async_copy_b128(gp, s_logit + e);
    }
    wait_async_all();
  }
  __syncthreads();

  // ---- exact per-row 1-D signed distance (cummax/cummin semantics) ----
  if (tid < ROWS) {
    const int* trow = s_tgt + tid * NW;
    int lastFG = -BIGD, lastBG = -BIGD;
    for (int i = 0; i < NW; ++i) {
      if (trow[i] > 0) lastFG = i; else lastBG = i;
      s_scni[tid * NW + i] = ((lastFG + BIGD) << 16) | (lastBG + BIGD);
    }
    int nextFG = BIGD, nextBG = BIGD;
    for (int i = NW - 1; i >= 0; --i) {
      if (trow[i] > 0) nextFG = i; else nextBG = i;
      int pk  = s_scni[tid * NW + i];
      int lFG = (pk >> 16) - BIGD;
      int lBG = (pk & 0xFFFF) - BIGD;
      int dO  = min(i - lFG, nextFG - i);   // dist to nearest foreground
      int dI  = min(i - lBG, nextBG - i);   // dist to nearest background
      s_scnf[tid * NW + i] = (float)(dO - dI);
    }
  }
  __syncthreads();

  // accumulator slots: [0..4] intersect, [5..9] psum, [10..14] volume,
  // [15] boundary, [16] ce@1, [17] ce@2, [18] ce@4
  float acc[19];
#pragma unroll
  for (int s = 0; s < 19; ++s) acc[s] = 0.0f;

  // ---- per-pixel softmax: dice partials + boundary + CE scale 1 ----
#pragma unroll
  for (int p = 0; p < (ROWS * NW) / TPB; ++p) {
    int pix = p * TPB + tid;
    int r = pix >> 9, w = pix & (NW - 1);
    float x[NC];
#pragma unroll
    for (int c = 0; c < NC; ++c) x[c] = s_logit[c * (ROWS * NW) + r * NW + w];
    int t = s_tgt[r * NW + w];
    float m = x[0];
#pragma unroll
    for (int c = 1; c < NC; ++c) m = fmaxf(m, x[c]);
    float ex[NC], Z = 0.0f;
#pragma unroll
    for (int c = 0; c < NC; ++c) { ex[c] = expf(x[c] - m); Z += ex[c]; }
    float invZ = 1.0f / Z, logZ = logf(Z);
    float xt = 0.0f;
#pragma unroll
    for (int c = 0; c < NC; ++c) {
      float pc  = ex[c] * invZ;
      float hit = (t == c) ? 1.0f : 0.0f;
      acc[c]      += pc * hit;
      acc[5 + c]  += pc;
      acc[10 + c] += hit;
      xt += hit * x[c];
    }
    acc[15] += (1.0f - ex[0] * invZ) * s_scnf[r * NW + w];
    acc[16] += (m + logZ - xt);   // -log p[t]
  }

  // ---- pooled CE, scale 2 (2x2 blocks) ----
  for (int blk = tid; blk < (ROWS / 2) * (NW / 2); blk += TPB) {
    int r0 = (blk >> 8) * 2, w0 = (blk & 255) * 2;
    float x[NC];
#pragma unroll
    for (int c = 0; c < NC; ++c) {
      const float* L = s_logit + c * (ROWS * NW);
      x[c] = 0.25f * (L[r0 * NW + w0] + L[r0 * NW + w0 + 1] +
                      L[(r0 + 1) * NW + w0] + L[(r0 + 1) * NW + w0 + 1]);
    }
    int ts = s_tgt[r0 * NW + w0] + s_tgt[r0 * NW + w0 + 1] +
             s_tgt[(r0 + 1) * NW + w0] + s_tgt[(r0 + 1) * NW + w0 + 1];
    int dt = ts >> 2;   // trunc(mean) of non-negative ints
    float m = x[0];
#pragma unroll
    for (int c = 1; c < NC; ++c) m = fmaxf(m, x[c]);
    float Z = 0.0f, xt = 0.0f;
#pragma unroll
    for (int c = 0; c < NC; ++c) { Z += expf(x[c] - m); xt += (dt == c) ? x[c] : 0.0f; }
    acc[17] += (m + logf(Z) - xt);
  }

  // ---- pooled CE, scale 4 (4x4 blocks == full tile height) ----
  if (tid < NW / 4) {
    int w0 = tid * 4;
    float x[NC];
    int ts = 0;
#pragma unroll
    for (int c = 0; c < NC; ++c) {
      const float* L = s_logit + c * (ROWS * NW);
      float s = 0.0f;
#pragma unroll
      for (int r = 0; r < 4; ++r)
#pragma unroll
        for (int dw = 0; dw < 4; ++dw) s += L[r * NW + w0 + dw];
      x[c] = s * (1.0f / 16.0f);
    }
#pragma unroll
    for (int r = 0; r < 4; ++r)
#pragma unroll
      for (int dw = 0; dw < 4; ++dw) ts += s_tgt[r * NW + w0 + dw];
    int dt = ts >> 4;
    float m = x[0];
#pragma unroll
    for (int c = 1; c < NC; ++c) m = fmaxf(m, x[c]);
    float Z = 0.0f, xt = 0.0f;
#pragma unroll
    for (int c = 0; c < NC; ++c) { Z += expf(x[c] - m); xt += (dt == c) ? x[c] : 0.0f; }
    acc[18] += (m + logf(Z) - xt);
  }

  __syncthreads();   // all reads of s_logit complete; safe to overwrite with s_acc

  {
    float* row = s_acc + tid * 33;
#pragma unroll
    for (int s = 0; s < 19; ++s) row[s] = acc[s];
#pragma unroll
    for (int s = 19; s < 32; ++s) row[s] = 0.0f;
  }
  __syncthreads();

#if defined(__HIP_DEVICE_COMPILE__) && __has_builtin(__builtin_amdgcn_wmma_f32_16x16x4_f32)
  // Matrix-engine block reduction: D = A(16x4 partials) x ones(4x16) + C.
  // Every wave runs uniformly (EXEC all-ones requirement); lanes 0/16 of wave 0
  // own the global atomics. A layout: lanes 0-15 hold K=0/1, lanes 16-31 K=2/3.
  {
    int m = tid & 15, half = (tid >> 4) & 1;
    v2f ones; ones.x = 1.0f; ones.y = 1.0f;
    v8f c0 = {}; v8f c1 = {};
    for (int chunk = 0; chunk < TPB / 4; ++chunk) {
      const float* r0 = s_acc + (chunk * 4 + half * 2 + 0) * 33;
      const float* r1 = s_acc + (chunk * 4 + half * 2 + 1) * 33;
      v2f a0, a1;
      a0.x = r0[m];      a0.y = r1[m];        // slots 0..15
      a1.x = r0[16 + m]; a1.y = r1[16 + m];   // slots 16..31 (19..31 are zero)
      c0 = __builtin_amdgcn_wmma_f32_16x16x4_f32(false, a0, false, ones,
                                                 (short)0, c0, false, false);
      c1 = __builtin_amdgcn_wmma_f32_16x16x4_f32(false, a1, false, ones,
                                                 (short)0, c1, false, false);
    }
    // D layout: VGPR r, lanes 0-15 -> M=r, lanes 16-31 -> M=r+8
    if (tid == 0) {
#pragma unroll
      for (int r = 0; r < 8; ++r) {
        atomicAdd(&ws[r], c0[r]);
        if (16 + r < 19) atomicAdd(&ws[16 + r], c1[r]);
      }
    }
    if (tid == 16) {
#pragma unroll
      for (int r = 0; r < 8; ++r) atomicAdd(&ws[8 + r], c0[r]);
    }
  }
#else
  if (tid < 19) {
    float s = 0.0f;
    for (int t = 0; t < TPB; ++t) s += s_acc[t * 33 + tid];
    atomicAdd(&ws[tid], s);
  }
#endif
}

__global__ void zero_ws_kernel(float* __restrict__ ws) {
  if (threadIdx.x < 32) ws[threadIdx.x] = 0.0f;
}

__global__ void finalize_kernel(const float* __restrict__ ws, float* __restrict__ out) {
  if (threadIdx.x != 0) return;
  float num = 0.0f, den = 0.0f;
#pragma unroll
  for (int c = 0; c < NC; ++c) {
    float vol = ws[10 + c];
    float wc  = 1.0f / (vol * vol + 1e-6f);
    num += wc * ws[c];
    den += wc * (ws[5 + c] + vol);
  }
  float dice = 1.0f - 2.0f * num / (den + 1e-6f);
  const float invN1 = 1.0f / (float)(NB * NH * NW);
  const float invN2 = 1.0f / (float)(NB * (NH / 2) * (NW / 2));
  const float invN4 = 1.0f / (float)(NB * (NH / 4) * (NW / 4));
  float boundary = ws[15] * invN1;
  float ce1 = ws[16] * invN1;
  float ce2 = ws[17] * invN2;
  float ce4 = ws[18] * invN4;
  float mscw = 0.5f * ce1 + 0.3f * ce2 + 0.2f * ce4;
  out[0] = dice + 1.0f * boundary + 0.5f * mscw;
  out[1] = dice;
  out[2] = boundary;
  out[3] = mscw;
}

extern "C" void kernel_launch(void* const* d_in, const int* in_sizes, int n_in,
                              void* d_out, int out_size, void* d_ws, size_t ws_size,
                              hipStream_t stream) {
  const float* pred   = (const float*)d_in[0];
  const int*   target = (const int*)d_in[1];
  float* out = (float*)d_out;
  float* ws  = (float*)d_ws;
  (void)in_sizes; (void)n_in; (void)out_size; (void)ws_size;

  zero_ws_kernel<<<1, 32, 0, stream>>>(ws);
  combined_loss_tile<<<NB * (NH / ROWS), TPB, 0, stream>>>(pred, target, ws);
  finalize_kernel<<<1, 32, 0, stream>>>(ws, out);
}